// CombinedPointCloudLoss_89764816486826
// MI455X (gfx1250) — compile-verified
//
#include <hip/hip_runtime.h>
#include <hip/hip_bf16.h>

// ext-vector types for the f32 WMMA builtin:
// A (16x4 f32)  = 64 vals / 32 lanes = 2 VGPRs -> v2f
// B (4x16 f32)  = 64 vals / 32 lanes = 2 VGPRs -> v2f
// C/D (16x16)   = 256 vals / 32 lanes = 8 VGPRs -> v8f
typedef __attribute__((ext_vector_type(2))) float v2f;
typedef __attribute__((ext_vector_type(8))) float v8f;

#define CHAMFER_W 1.0f
#define INTENSITY_W 0.5f

// ---------------------------------------------------------------------------
// Kernel 0: pack each point array into the exact WMMA B-operand layout:
//   lanes 0-15  (K0,K1): (-2x, -2y)   -> ref[i]
//   lanes 16-31 (K2,K3): (-2z, |r|^2) -> ref[i + n]
// With A's K3 slot set to 1.0, D = A*B + 0 = -2 q.r + |r|^2 directly.
// ---------------------------------------------------------------------------
__global__ __launch_bounds__(256) void prep_kernel(
    const float4* __restrict__ pred, const float4* __restrict__ targ,
    float2* __restrict__ refP, float2* __restrict__ refT, int n)
{
  const int i = blockIdx.x * blockDim.x + threadIdx.x;
  if (i >= n) return;
  const float4 p = pred[i];
  refP[i]     = make_float2(-2.0f * p.x, -2.0f * p.y);
  refP[i + n] = make_float2(-2.0f * p.z, p.x * p.x + p.y * p.y + p.z * p.z);
  const float4 t = targ[i];
  refT[i]     = make_float2(-2.0f * t.x, -2.0f * t.y);
  refT[i + n] = make_float2(-2.0f * t.z, t.x * t.x + t.y * t.y + t.z * t.z);
}

// ---------------------------------------------------------------------------
// Kernel 1: exact 1-NN via V_WMMA_F32_16X16X4_F32.
// One wave per 16-row query tile; sweeps all ref tiles. Inner loop:
//   1 x global_load_b64 (pre-packed B pair) + 1 x WMMA (C = inline 0)
//   + 8 cmp + 16 cndmask running argmin  (the irreducible bookkeeping floor).
// C/D layout: VGPR j, lanes 0-15 -> row M=j, lanes 16-31 -> row M=j+8; col = lane&15.
// ---------------------------------------------------------------------------
__global__ __launch_bounds__(256) void nn_wmma_kernel(
    const float4* __restrict__ pred, const float4* __restrict__ targ,
    const float2* __restrict__ refP, const float2* __restrict__ refT,
    int* __restrict__ idx_p2t, int* __restrict__ idx_t2p, int n)
{
  const int lane   = threadIdx.x & 31;
  const int wave   = threadIdx.x >> 5;
  const int ntiles = n >> 4;                       // 16-point tiles per direction
  const int tile   = blockIdx.x * (blockDim.x >> 5) + wave;
  if (tile >= 2 * ntiles) return;                  // wave-uniform (grid sized exactly)

  const float4* q; const float2* rpk; int* outIdx; int qt;
  if (tile < ntiles) { q = pred; rpk = refT; outIdx = idx_p2t; qt = tile; }
  else               { q = targ; rpk = refP; outIdx = idx_t2p; qt = tile - ntiles; }

  const int  col = lane & 15;
  const bool hi  = lane >= 16;

  // A matrix: lane (and lane+16) own query row qt*16+col.
  //   lanes 0-15:  (K0,K1) = (x, y)
  //   lanes 16-31: (K2,K3) = (z, 1.0)   (K3=1 picks up |r|^2 from B's K3 row)
  const float4 qp = q[qt * 16 + col];
  v2f A;
  A.x = hi ? qp.z : qp.x;
  A.y = hi ? 1.0f : qp.y;

  // Per-lane base into the packed B stream: hi lanes read the (K2,K3) half.
  const float2* rlane = rpk + (hi ? n : 0);

  float best[8];
  int   bidx[8];
#pragma unroll
  for (int j = 0; j < 8; ++j) { best[j] = 3.0e38f; bidx[j] = 0; }

#pragma unroll 4
  for (int t = 0; t < ntiles; ++t) {
    const int nidx = t * 16 + col;                 // this lane's ref column
    const v2f B    = *(const v2f*)(rlane + nidx);  // single b64 load, zero VALU prep

    v8f C = {};                                    // lowers to inline SRC2 = 0
    v8f D = __builtin_amdgcn_wmma_f32_16x16x4_f32(
        /*neg_a=*/false, A, /*neg_b=*/false, B,
        /*c_mod=*/(short)0, C, /*reuse_a=*/false, /*reuse_b=*/false);

#pragma unroll
    for (int j = 0; j < 8; ++j) {                  // running min/argmin (cndmask only)
      const bool c = D[j] < best[j];
      best[j] = c ? D[j] : best[j];
      bidx[j] = c ? nidx : bidx[j];
    }
  }

  // Cross-lane argmin: butterfly within each 16-lane half (rows j and j+8).
#pragma unroll
  for (int j = 0; j < 8; ++j) {
    float d  = best[j];
    int   bi = bidx[j];
#pragma unroll
    for (int m = 1; m < 16; m <<= 1) {
      const float od = __shfl_xor(d,  m, 32);
      const int   oi = __shfl_xor(bi, m, 32);
      const bool take = (od < d) || (od == d && oi < bi);   // tie -> smaller index
      d  = take ? od : d;
      bi = take ? oi : bi;
    }
    if (col == 0) {                                // lane 0 -> row j, lane 16 -> row j+8
      outIdx[qt * 16 + (hi ? j + 8 : j)] = bi;
    }
  }
}

// ---------------------------------------------------------------------------
// Kernel 2: per-element loss terms + deterministic block reduction.
// Threads [0,n): pred->target chamfer + intensity; threads [n,2n): target->pred.
// ---------------------------------------------------------------------------
__global__ __launch_bounds__(256) void loss_partial_kernel(
    const float4* __restrict__ pred, const float4* __restrict__ targ,
    const int* __restrict__ idx_p2t, const int* __restrict__ idx_t2p,
    float* __restrict__ partial, int n)
{
  __shared__ float sm[256];
  const int   i    = blockIdx.x * blockDim.x + threadIdx.x;
  const float invn = 1.0f / (float)n;
  float contrib = 0.0f;

  if (i < n) {
    const float4 p  = pred[i];
    const float4 tg = targ[idx_p2t[i]];
    const float dx = p.x - tg.x, dy = p.y - tg.y, dz = p.z - tg.z;
    const float dn = sqrtf(dx * dx + dy * dy + dz * dz);
    const float di = p.w - tg.w;
    contrib = (CHAMFER_W * dn + INTENSITY_W * di * di) * invn;
  } else if (i < 2 * n) {
    const int    j  = i - n;
    const float4 tg = targ[j];
    const float4 p  = pred[idx_t2p[j]];
    const float dx = tg.x - p.x, dy = tg.y - p.y, dz = tg.z - p.z;
    contrib = CHAMFER_W * sqrtf(dx * dx + dy * dy + dz * dz) * invn;
  }

  sm[threadIdx.x] = contrib;
  __syncthreads();
#pragma unroll
  for (int s = 128; s > 0; s >>= 1) {
    if ((int)threadIdx.x < s) sm[threadIdx.x] += sm[threadIdx.x + s];
    __syncthreads();
  }
  if (threadIdx.x == 0) partial[blockIdx.x] = sm[0];
}

// ---------------------------------------------------------------------------
// Kernel 3: fixed-order serial sum of partials -> scalar loss (deterministic).
// ---------------------------------------------------------------------------
__global__ void finalize_kernel(const float* __restrict__ partial, int nparts,
                                float* __restrict__ out)
{
  if (threadIdx.x == 0 && blockIdx.x == 0) {
    float s = 0.0f;
    for (int k = 0; k < nparts; ++k) s += partial[k];
    out[0] = s;
  }
}

extern "C" void kernel_launch(void* const* d_in, const int* in_sizes, int n_in,
                              void* d_out, int out_size, void* d_ws, size_t ws_size,
                              hipStream_t stream) {
  const float4* pred = (const float4*)d_in[0];   // [32768,4] f32
  const float4* targ = (const float4*)d_in[1];   // [32768,4] f32
  const int n = in_sizes[0] / 4;                 // 32768 points

  // Scratch layout:
  //   [idx_p2t: n ints][idx_t2p: n ints]
  //   [refP: 2n float2][refT: 2n float2]
  //   [partials: floats]
  int*    idx_p2t = (int*)d_ws;
  int*    idx_t2p = idx_p2t + n;
  float2* refP    = (float2*)(idx_t2p + n);
  float2* refT    = refP + 2 * n;
  float*  partial = (float*)(refT + 2 * n);

  // Pack B operands (one pass over both arrays; ~1 MB, L2-resident).
  const int prepBlocks = (n + 255) / 256;
  prep_kernel<<<prepBlocks, 256, 0, stream>>>(pred, targ, refP, refT, n);

  // NN search: 2*(n/16) query tiles, 8 waves (256 threads) per block.
  const int ntiles     = n / 16;
  const int totalWaves = 2 * ntiles;
  const int nnBlocks   = (totalWaves + 7) / 8;
  nn_wmma_kernel<<<nnBlocks, 256, 0, stream>>>(pred, targ, refP, refT,
                                               idx_p2t, idx_t2p, n);

  // Loss terms + block reduction.
  const int lossBlocks = (2 * n + 255) / 256;
  loss_partial_kernel<<<lossBlocks, 256, 0, stream>>>(pred, targ, idx_p2t, idx_t2p,
                                                      partial, n);

  // Deterministic final sum into d_out[0].
  finalize_kernel<<<1, 32, 0, stream>>>(partial, lossBlocks, (float*)d_out);
}